// NeuralODEClassifier_44100724195482
// MI455X (gfx1250) — compile-verified
//
#include <hip/hip_runtime.h>
#include <hip/hip_bf16.h>
#include <math.h>

// ext-vector types matching gfx1250 WMMA f32 16x16x4 operand shapes:
//   A: 16x4 f32  -> 64 vals / 32 lanes -> 2 VGPRs
//   B: 4x16 f32  -> 2 VGPRs
//   C/D: 16x16   -> 8 VGPRs
typedef float v2f __attribute__((ext_vector_type(2)));
typedef float v8f __attribute__((ext_vector_type(8)));

#if defined(__HIP_DEVICE_COMPILE__)
#if __has_builtin(__builtin_amdgcn_tanhf)
#define FAST_TANH(x) __builtin_amdgcn_tanhf(x)
#elif __has_builtin(__builtin_amdgcn_tanh_f32)
#define FAST_TANH(x) __builtin_amdgcn_tanh_f32(x)
#else
#define FAST_TANH(x) tanhf(x)
#endif
#if __has_builtin(__builtin_amdgcn_fmed3f)
#define RELU1(x) __builtin_amdgcn_fmed3f((x), 0.0f, __builtin_inff())
#else
#define RELU1(x) ((x) > 0.0f ? (x) : 0.0f)
#endif
#if !__has_builtin(__builtin_amdgcn_wmma_f32_16x16x4_f32)
#error "missing __builtin_amdgcn_wmma_f32_16x16x4_f32 for gfx1250 (device pass)"
#endif
#else
// host pass: body is never codegen'd, just needs to parse
#define FAST_TANH(x) tanhf(x)
#define RELU1(x) ((x) > 0.0f ? (x) : 0.0f)
#endif

static __device__ __forceinline__ v8f wmma_f32x4(v2f a, v2f b, v8f c) {
  // 8 args: (neg_a, A, neg_b, B, c_mod, C, reuse_a, reuse_b)
  return __builtin_amdgcn_wmma_f32_16x16x4_f32(false, a, false, b, (short)0, c,
                                               false, false);
}

static __device__ __forceinline__ v8f splat8(float x) {
  v8f r;
#pragma unroll
  for (int i = 0; i < 8; ++i) r[i] = x;
  return r;
}

static __device__ __forceinline__ v8f relu8(v8f x) {
  v8f r;
#pragma unroll
  for (int i = 0; i < 8; ++i) r[i] = RELU1(x[i]);
  return r;
}

#define BLK 256

__global__ __launch_bounds__(BLK) void neural_ode_cls_kernel(
    const float* __restrict__ y0in, const float* __restrict__ tg,
    const float* __restrict__ W1, const float* __restrict__ b1,
    const float* __restrict__ W2, const float* __restrict__ b2,
    const float* __restrict__ Wc1, const float* __restrict__ bc1,
    const float* __restrict__ Wc2, const float* __restrict__ bc2,
    const float* __restrict__ Wc3, const float* __restrict__ bc3,
    float* __restrict__ out, int Bn, int Tn) {
  __shared__ float ybuf[BLK][2];
  // per-wave 16x64 transpose scratch, padded to 68 cols to spread banks
  __shared__ float hbuf[8][16][68];

  const int tid = threadIdx.x;
  const int gid = blockIdx.x * BLK + tid;
  const int lane = tid & 31;
  const int wave = tid >> 5;
  const int l = lane & 15;    // column / row-within-halftile index
  const int half = lane >> 4; // 0: lanes 0-15, 1: lanes 16-31

  // ---------------- Phase 1: forward Euler, one 2-d state per lane ----------
  // Skinny (K=2,N=2) matmuls: per-lane VALU + v_tanh_f32 beats padded WMMA.
  // Weights are wave-uniform; fetch them in 16-wide chunks per step so they
  // stay SMEM/K$-resident instead of bloating the wave's VGPR footprint.
  float ya = 0.f, yb = 0.f;
  if (gid < Bn) {
    ya = y0in[2 * gid + 0];
    yb = y0in[2 * gid + 1];
  }
#pragma unroll 1
  for (int s = 0; s + 1 < Tn; ++s) {
    const float dt = tg[s + 1] - tg[s];
    float f0 = b2[0], f1 = b2[1];
#pragma unroll 1
    for (int c = 0; c < 4; ++c) {
      const float* __restrict__ w1a = W1 + 16 * c;       // row 0 of W1
      const float* __restrict__ w1b = W1 + 64 + 16 * c;  // row 1 of W1
      const float* __restrict__ bb = b1 + 16 * c;
      const float* __restrict__ w2 = W2 + 32 * c;
#pragma unroll
      for (int j = 0; j < 16; ++j) {
        const float h = FAST_TANH(fmaf(yb, w1b[j], fmaf(ya, w1a[j], bb[j])));
        f0 = fmaf(h, w2[2 * j + 0], f0);
        f1 = fmaf(h, w2[2 * j + 1], f1);
      }
    }
    ya = fmaf(dt, f0, ya);
    yb = fmaf(dt, f1, yb);
  }
  ybuf[tid][0] = ya;
  ybuf[tid][1] = yb;
  __syncthreads();

  // ---------------- Phase 2: classifier via V_WMMA_F32_16X16X4_F32 ----------
  // B-operand layout (4x16): VGPR0 = {K0 | K2} across lane halves, VGPR1 = {K1 | K3}.
  // A-operand layout (16x4): VGPR0 = {K0 | K2}, VGPR1 = {K1 | K3}, row M = l.
  // C/D layout: VGPR v -> row v (half 0) / v+8 (half 1), col = l.
  v2f wb1[4];
  float cb1[4], cb2[4];
#pragma unroll
  for (int n = 0; n < 4; ++n) {
    const float wx = Wc1[0 * 64 + 16 * n + l];
    const float wy = Wc1[1 * 64 + 16 * n + l];
    wb1[n].x = half ? 0.f : wx; // K=0 row (K=2 pad -> 0)
    wb1[n].y = half ? 0.f : wy; // K=1 row (K=3 pad -> 0)
    cb1[n] = bc1[16 * n + l];
    cb2[n] = bc2[16 * n + l];
  }
  const int lc = (l < 3) ? l : 0; // clamped, always in-bounds column for Wc3
  const float cb3 = (l < 3) ? bc3[lc] : 0.f;

#pragma unroll 1
  for (int tile = 0; tile < 2; ++tile) {
    const int r0 = 32 * wave + 16 * tile; // block-local batch-row base

    // --- GEMM1: h1[16,64] = relu(y[16,2] @ Wc1[2,64] + bc1), K padded 2->4
    const float yt0 = ybuf[r0 + l][0];
    const float yt1 = ybuf[r0 + l][1];
    v2f ay;
    ay.x = half ? 0.f : yt0;
    ay.y = half ? 0.f : yt1;

    v8f h1[4];
#pragma unroll
    for (int n = 0; n < 4; ++n)
      h1[n] = relu8(wmma_f32x4(ay, wb1[n], splat8(cb1[n])));

    // transpose h1 (D layout) -> row-major LDS so it can be re-fed as A
    __syncthreads();
#pragma unroll
    for (int n = 0; n < 4; ++n)
#pragma unroll
      for (int v = 0; v < 8; ++v)
        hbuf[wave][v + 8 * half][16 * n + l] = h1[n][v];
    __syncthreads();

    // --- GEMM2: h2[16,64] = relu(h1 @ Wc2[64,64] + bc2); 16 K-chunks x 4 N-tiles
    // Modest unroll: keeps B-operand loads in flight without blowing VGPRs.
    v8f h2[4];
#pragma unroll
    for (int n = 0; n < 4; ++n) h2[n] = splat8(cb2[n]);
#pragma unroll 2
    for (int kc = 0; kc < 16; ++kc) {
      const v2f a = *(const v2f*)&hbuf[wave][l][4 * kc + 2 * half];
#pragma unroll
      for (int n = 0; n < 4; ++n) {
        v2f bw;
        bw.x = Wc2[(4 * kc + 2 * half + 0) * 64 + 16 * n + l];
        bw.y = Wc2[(4 * kc + 2 * half + 1) * 64 + 16 * n + l];
        h2[n] = wmma_f32x4(a, bw, h2[n]);
      }
    }
#pragma unroll
    for (int n = 0; n < 4; ++n) h2[n] = relu8(h2[n]);

    __syncthreads();
#pragma unroll
    for (int n = 0; n < 4; ++n)
#pragma unroll
      for (int v = 0; v < 8; ++v)
        hbuf[wave][v + 8 * half][16 * n + l] = h2[n][v];
    __syncthreads();

    // --- GEMM3: out[16,3] = h2 @ Wc3[64,3] + bc3 (N padded 3->16)
    v8f o = splat8(cb3);
#pragma unroll 4
    for (int kc = 0; kc < 16; ++kc) {
      const v2f a = *(const v2f*)&hbuf[wave][l][4 * kc + 2 * half];
      const float w0 = Wc3[(4 * kc + 2 * half + 0) * 3 + lc];
      const float w1 = Wc3[(4 * kc + 2 * half + 1) * 3 + lc];
      v2f bw;
      bw.x = (l < 3) ? w0 : 0.f;
      bw.y = (l < 3) ? w1 : 0.f;
      o = wmma_f32x4(a, bw, o);
    }

    if (l < 3) {
      const int rowbase = blockIdx.x * BLK + r0 + 8 * half;
#pragma unroll
      for (int v = 0; v < 8; ++v) {
        const int gr = rowbase + v;
        if (gr < Bn) out[gr * 3 + l] = o[v];
      }
    }
  }
}

extern "C" void kernel_launch(void* const* d_in, const int* in_sizes, int n_in,
                              void* d_out, int out_size, void* d_ws,
                              size_t ws_size, hipStream_t stream) {
  const float* y0 = (const float*)d_in[0];
  const float* t = (const float*)d_in[1];
  const float* W1 = (const float*)d_in[2];
  const float* b1 = (const float*)d_in[3];
  const float* W2 = (const float*)d_in[4];
  const float* b2 = (const float*)d_in[5];
  const float* Wc1 = (const float*)d_in[6];
  const float* bc1 = (const float*)d_in[7];
  const float* Wc2 = (const float*)d_in[8];
  const float* bc2 = (const float*)d_in[9];
  const float* Wc3 = (const float*)d_in[10];
  const float* bc3 = (const float*)d_in[11];

  const int Bn = in_sizes[0] / 2;
  const int Tn = in_sizes[1];
  const int grid = (Bn + BLK - 1) / BLK;

  neural_ode_cls_kernel<<<grid, BLK, 0, stream>>>(
      y0, t, W1, b1, W2, b2, Wc1, bc1, Wc2, bc2, Wc3, bc3, (float*)d_out, Bn,
      Tn);
}